// RandomGaussianBlur_15006615733005
// MI455X (gfx1250) — compile-verified
//
#include <hip/hip_runtime.h>

typedef __attribute__((ext_vector_type(2))) float v2f;
typedef __attribute__((ext_vector_type(4))) float v4f;
typedef __attribute__((ext_vector_type(8))) float v8f;

#define TILE   16
#define HALO   5
#define INW    26           // TILE + 2*HALO
#define NKB    7            // K padded 26 -> 28, 7 blocks of 4

#define DIM_D  160
#define DIM_H  192
#define DIM_W  192
#define NBATCH 4

#define P1ROWS (INW * INW)          // 676 real rows after pass 1
#define P1PAD  (((P1ROWS + 15) / 16) * 16)   // 688, full 16-row groups (slack => no store guards)

// Fused separable 3D Gaussian blur. One workgroup = one 16^3 output tile.
// Each 1D pass is a banded matmul (rows x 26) * G(26x16) done with chained
// V_WMMA_F32_16X16X4_F32 (K padded to 28). Final tile is staged in LDS and
// written out with coalesced float4 stores.
__global__ __launch_bounds__(256)
void gauss3d_wmma_kernel(const float* __restrict__ x, float* __restrict__ out)
{
    __shared__ float sIn[INW * INW * INW + 8];   // 17584 floats, input tile w/ halo
    __shared__ float sT1[P1PAD * TILE];          // 11008 floats, after W-blur (+slack rows)
    float* sT2  = sIn;                           // 6656 floats, aliases sIn (dead after pass 1)
    float* sOut = sT1;                           // 4096 floats, aliases sT1 (dead after pass 2)

    const int tid  = threadIdx.x;
    const int lane = tid & 31;
    const int wave = tid >> 5;
    const int nIdx = lane & 15;   // A-matrix row (compute) / D-matrix column N (store)
    const int khi  = lane >> 4;   // K half-select per documented f32 A layout

    const int wT = blockIdx.x;
    const int hT = blockIdx.y;
    const int zT = blockIdx.z;
    const int dT = zT % (DIM_D / TILE);
    const int b  = zT / (DIM_D / TILE);

    const int d0 = dT * TILE, h0 = hT * TILE, w0 = wT * TILE;
    const size_t HW   = (size_t)DIM_H * DIM_W;
    const size_t bOff = (size_t)b * DIM_D * HW;

    // Gaussian taps, closed form (sigma = 1.5, 11 taps, normalized)
    const float cexp = 1.0f / (2.0f * 1.5f * 1.5f);
    float wsum = 0.0f;
    #pragma unroll
    for (int t = 0; t < 11; ++t) { float d = (float)(t - 5); wsum += __expf(-d * d * cexp); }
    const float winv = 1.0f / wsum;

    // Band matrix fragments: B[k][n] = w[k-n] if 0<=k-n<=10 else 0 (k=26,27 pad -> 0).
    // B layout mirrors A: VGPR0 holds K=kb*4+2*khi row, VGPR1 the next K row.
    v2f bf[NKB];
    #pragma unroll
    for (int kb = 0; kb < NKB; ++kb) {
        int k0 = kb * 4 + 2 * khi;
        int t0 = k0 - nIdx;
        int t1 = k0 + 1 - nIdx;
        float f0 = 0.0f, f1 = 0.0f;
        if (t0 >= 0 && t0 < 11) { float d = (float)(t0 - 5); f0 = __expf(-d * d * cexp) * winv; }
        if (t1 >= 0 && t1 < 11) { float d = (float)(t1 - 5); f1 = __expf(-d * d * cexp) * winv; }
        bf[kb].x = f0;
        bf[kb].y = f1;
    }

    // ---- stage 0: cooperative load of 26^3 halo tile (zero padding outside) ----
    for (int i = tid; i < INW * INW * INW; i += 256) {
        int dz = i / (INW * INW);
        int r  = i % (INW * INW);
        int dy = r / INW;
        int dx = r % INW;
        int gd = d0 + dz - HALO, gh = h0 + dy - HALO, gw = w0 + dx - HALO;
        float v = 0.0f;
        if (gd >= 0 && gd < DIM_D && gh >= 0 && gh < DIM_H && gw >= 0 && gw < DIM_W)
            v = x[bOff + (size_t)gd * HW + (size_t)gh * DIM_W + gw];
        sIn[i] = v;
    }
    __syncthreads();

    // ---- pass 1: blur along W. rows = (d,h) in [0,676), K stride 1.
    //      Rows clamped to 675; duplicate results land in sT1 slack rows (no guards). ----
    {
        const int nGrp = P1PAD / 16;            // 43
        for (int g = wave; g < nGrp; g += 8) {
            int rowA = g * 16 + nIdx;
            if (rowA > P1ROWS - 1) rowA = P1ROWS - 1;
            const int rowOff = rowA * INW;
            v8f acc = {};
            #pragma unroll
            for (int kb = 0; kb < NKB; ++kb) {
                int k0  = kb * 4 + 2 * khi;
                int ka  = (k0     <= INW - 1) ? k0     : INW - 1;  // B is 0 there anyway
                int kb1 = (k0 + 1 <= INW - 1) ? k0 + 1 : INW - 1;
                v2f a;
                a.x = sIn[rowOff + ka];
                a.y = sIn[rowOff + kb1];
                acc = __builtin_amdgcn_wmma_f32_16x16x4_f32(
                        false, a, false, bf[kb], (short)0, acc, false, false);
            }
            #pragma unroll
            for (int rr = 0; rr < 8; ++rr) {
                int m = g * 16 + rr + 8 * khi;
                sT1[m * TILE + nIdx] = acc[rr];   // unconditional: slack rows absorb m>=676
            }
        }
    }
    __syncthreads();

    // ---- pass 2: blur along H. src sT1[d][h][w'] (26x26x16); rows = (d,w'), K stride 16 ----
    {
        const int nGrp = (INW * TILE) / 16;     // 26
        for (int g = wave; g < nGrp; g += 8) {
            int rowA = g * 16 + nIdx;           // d*16 + w'
            const int rowOff = (rowA >> 4) * (INW * TILE) + (rowA & 15);
            v8f acc = {};
            #pragma unroll
            for (int kb = 0; kb < NKB; ++kb) {
                int k0  = kb * 4 + 2 * khi;
                int ka  = (k0     <= INW - 1) ? k0     : INW - 1;
                int kb1 = (k0 + 1 <= INW - 1) ? k0 + 1 : INW - 1;
                v2f a;
                a.x = sT1[rowOff + ka  * TILE];
                a.y = sT1[rowOff + kb1 * TILE];
                acc = __builtin_amdgcn_wmma_f32_16x16x4_f32(
                        false, a, false, bf[kb], (short)0, acc, false, false);
            }
            #pragma unroll
            for (int rr = 0; rr < 8; ++rr) {
                int m = g * 16 + rr + 8 * khi;  // d*16 + w'
                sT2[(m >> 4) * (TILE * TILE) + nIdx * TILE + (m & 15)] = acc[rr];
            }
        }
    }
    __syncthreads();

    // ---- pass 3: blur along D. src sT2[d][h'][w'] (26x16x16); rows = (h',w'), K stride 256.
    //      Results staged in LDS out-tile [d'][h'][w'] for coalesced write-out. ----
    {
        const int nGrp = (TILE * TILE) / 16;    // 16
        for (int g = wave; g < nGrp; g += 8) {
            int rowA = g * 16 + nIdx;           // h'*16 + w'
            v8f acc = {};
            #pragma unroll
            for (int kb = 0; kb < NKB; ++kb) {
                int k0  = kb * 4 + 2 * khi;
                int ka  = (k0     <= INW - 1) ? k0     : INW - 1;
                int kb1 = (k0 + 1 <= INW - 1) ? k0 + 1 : INW - 1;
                v2f a;
                a.x = sT2[rowA + ka  * (TILE * TILE)];
                a.y = sT2[rowA + kb1 * (TILE * TILE)];
                acc = __builtin_amdgcn_wmma_f32_16x16x4_f32(
                        false, a, false, bf[kb], (short)0, acc, false, false);
            }
            #pragma unroll
            for (int rr = 0; rr < 8; ++rr) {
                int m = g * 16 + rr + 8 * khi;  // h'*16 + w'
                sOut[nIdx * (TILE * TILE) + m] = acc[rr];   // [d'][h'][w']
            }
        }
    }
    __syncthreads();

    // ---- stage 4: coalesced float4 write-out (each wave streams contiguous 512B) ----
    {
        const v4f* sOut4 = (const v4f*)sOut;
        for (int j = tid; j < (TILE * TILE * TILE) / 4; j += 256) {
            int e  = j * 4;                     // element index, w' multiple of 4
            int dd = e >> 8;
            int hh = (e >> 4) & 15;
            int ww = e & 15;
            v4f* dst = (v4f*)&out[bOff + (size_t)(d0 + dd) * HW
                                  + (size_t)(h0 + hh) * DIM_W + (w0 + ww)];
            *dst = sOut4[j];
        }
    }
}

extern "C" void kernel_launch(void* const* d_in, const int* in_sizes, int n_in,
                              void* d_out, int out_size, void* d_ws, size_t ws_size,
                              hipStream_t stream) {
    (void)in_sizes; (void)n_in; (void)out_size; (void)d_ws; (void)ws_size;
    const float* x = (const float*)d_in[0];
    float* out     = (float*)d_out;
    dim3 grid(DIM_W / TILE, DIM_H / TILE, (DIM_D / TILE) * NBATCH);  // 12 x 12 x 40
    gauss3d_wmma_kernel<<<grid, dim3(256), 0, stream>>>(x, out);
}